// FullGaussianProjector_66984309948584
// MI455X (gfx1250) — compile-verified
//
#include <hip/hip_runtime.h>
#include <math.h>

typedef __attribute__((ext_vector_type(2))) float v2f;
typedef __attribute__((ext_vector_type(4))) float v4f;
typedef __attribute__((ext_vector_type(8))) float v8f;

#define NG      512
#define FULL    224
#define ESTR    336           // 21 groups * 16 = 336 slots (324 real halo points
                              // + 12 write-only pad); 336*4B row = 16B aligned
#define PSTRIDE 12

// ---------------------------------------------------------------------------
// Kernel 1: per-gaussian prep. One block of 512 threads (one thread/gaussian),
// block-wide min/max reductions for the screen mapping per view.
// params[v][n] = {a, 2b, c, alpha, beta, gamma, xy.x, xy.y, opacity*imp, pad x3}
// ---------------------------------------------------------------------------
__global__ __launch_bounds__(512) void prep_k(const float* __restrict__ pos,
                                              const float* __restrict__ cov,
                                              const float* __restrict__ opa,
                                              const float* __restrict__ impo,
                                              float* __restrict__ params)
{
    __shared__ float ra[512], rb[512], rc[512], rd[512];
    const int n = threadIdx.x;
    const float px = pos[n*3+0], py = pos[n*3+1], pz = pos[n*3+2];
    float c00 = cov[n*9+0], c01 = cov[n*9+1], c02 = cov[n*9+2];
    float c10 = cov[n*9+3], c11 = cov[n*9+4], c12 = cov[n*9+5];
    float c20 = cov[n*9+6], c21 = cov[n*9+7], c22 = cov[n*9+8];
    const float wsc = opa[n] * fminf(fmaxf(impo[n], 0.5f), 2.0f);

    for (int v = 0; v < 3; ++v) {
        // symmetrize, +EPS*I, Frobenius normalize (COV_HP = 1)
        float s01 = 0.5f*(c01+c10), s02 = 0.5f*(c02+c20), s12 = 0.5f*(c12+c21);
        c01 = c10 = s01; c02 = c20 = s02; c12 = c21 = s12;
        c00 += 1e-6f; c11 += 1e-6f; c22 += 1e-6f;
        float nrm = sqrtf(c00*c00 + c11*c11 + c22*c22 + 2.0f*(s01*s01 + s02*s02 + s12*s12));
        float sc = 1.0f / (nrm + 1e-6f);
        c00 *= sc; c11 *= sc; c22 *= sc;
        c01 *= sc; c10 = c01; c02 *= sc; c20 = c02; c12 *= sc; c21 = c12;
        // symmetric 3x3 inverse via adjugate
        float m00 = c11*c22 - c12*c12;
        float m01 = c02*c12 - c01*c22;
        float m02 = c01*c12 - c02*c11;
        float m11 = c00*c22 - c02*c02;
        float m12 = c01*c02 - c00*c12;
        float m22 = c00*c11 - c01*c01;
        float idet = 1.0f / (c00*m00 + c01*m01 + c02*m02);
        float i00 = m00*idet, i01 = m01*idet, i02 = m02*idet;
        float i11 = m11*idet, i12 = m12*idet, i22 = m22*idet;
        // J cinv J^T per view + 1e-10 on diagonal; projected coords
        float a, b, cc, qx, qy;
        if (v == 0)      { a = i00; b = i01; cc = i11; qx = px; qy = py; }
        else if (v == 1) { a = i00; b = i02; cc = i22; qx = px; qy = pz; }
        else             { a = i22; b = i12; cc = i11; qx = pz; qy = py; }
        a += 1e-10f; cc += 1e-10f;

        // block-wide min/max of projected coords over N
        ra[n] = qx; rb[n] = qx; rc[n] = qy; rd[n] = qy;
        __syncthreads();
        for (int s = 256; s > 0; s >>= 1) {
            if (n < s) {
                ra[n] = fminf(ra[n], ra[n+s]); rb[n] = fmaxf(rb[n], rb[n+s]);
                rc[n] = fminf(rc[n], rc[n+s]); rd[n] = fmaxf(rd[n], rd[n+s]);
            }
            __syncthreads();
        }
        float mnx = ra[0], mxx = rb[0], mny = rc[0], mxy = rd[0];
        __syncthreads();    // arrays reused next view

        float rngx = mxx - mnx + 1e-6f;
        float mn2x = mnx - 0.5f*rngx, mx2x = mxx + 0.5f*rngx;
        float r2x  = mx2x - mn2x + 1e-6f;
        float rngy = mxy - mny + 1e-6f;
        float mn2y = mny - 0.5f*rngy, mx2y = mxy + 0.5f*rngy;
        float r2y  = mx2y - mn2y + 1e-6f;
        float xyx = fminf(fmaxf((qx - mn2x)/r2x * 111.0f, 0.0f), 111.0f);
        float xyy = fminf(fmaxf((qy - mn2y)/r2y * 111.0f, 0.0f), 111.0f);

        // rank-6 factor:  e(g) = a gx^2 + 2b gx gy + c gy^2 + al gx + be gy + ga
        float al = -2.0f*(a*xyx + b*xyy);
        float be = -2.0f*(b*xyx + cc*xyy);
        float ga = a*xyx*xyx + 2.0f*b*xyx*xyy + cc*xyy*xyy;
        float* P = params + (v*NG + n)*PSTRIDE;
        P[0] = a;   P[1] = 2.0f*b; P[2] = cc;  P[3] = al; P[4] = be; P[5] = ga;
        P[6] = xyx; P[7] = xyy;    P[8] = wsc; P[9] = 0.f; P[10] = 0.f; P[11] = 0.f;
    }
}

// ---------------------------------------------------------------------------
// Kernel 2: main splat. grid=(7,7,3) tiles x views, 256 threads (8 waves).
// Each block: 16x16 half-res tile -> 32x32 output pixels.
// Per 16-gaussian chunk: WMMA (f32 16x16x4, rank 6 padded to 8) evaluates the
// quadratic form on an 18x18 halo of half-res points into LDS; then each
// thread bilinearly upsamples / clips / exps / masks 4 pixels x 16 gaussians,
// keeping running sum/min/max (the N-reduction collapses analytically).
// ---------------------------------------------------------------------------
__global__ __launch_bounds__(256) void splat_k(const float* __restrict__ params,
                                               float* __restrict__ dm)
{
    __shared__ float E[16*ESTR];   // [gauss 0..15][halo slot 0..335]
    __shared__ float P[16*PSTRIDE];

    const int t    = threadIdx.x;
    const int tx   = blockIdx.x, ty = blockIdx.y, v = blockIdx.z;
    const int wave = t >> 5, lane = t & 31;
    const int nl   = lane & 15;
    const bool lo16 = lane < 16;

    // per-pixel precompute (4 output pixels per thread)
    float S[4], wmn[4], wmx[4], wxl_[4], wyl_[4], mkx_[4], mky_[4];
    int   e00i[4], oidx[4];
#pragma unroll
    for (int i = 0; i < 4; ++i) {
        int p = t + 256*i;
        int I = p >> 5, J = p & 31;
        int Ig = ty*32 + I, Jg = tx*32 + J;
        int ky = Ig >> 1, kx = Jg >> 1;
        // bilinear x2 upsample: src s = i/2 - 0.25 (edge behavior == clamp)
        int ylo = (Ig & 1) ? ky : ky - 1;
        int xlo = (Jg & 1) ? kx : kx - 1;
        wyl_[i] = (Ig & 1) ? 0.75f : 0.25f;
        wxl_[i] = (Jg & 1) ? 0.75f : 0.25f;
        e00i[i] = (ylo - (ty*16 - 1))*18 + (xlo - (tx*16 - 1));
        mkx_[i] = (float)kx; mky_[i] = (float)ky;
        oidx[i] = (v*FULL + Ig)*FULL + Jg;
        S[i] = 0.0f; wmn[i] = 3.4e38f; wmx[i] = -3.4e38f;
    }
    const int hy0 = ty*16 - 1, hx0 = tx*16 - 1;
    const float clxl = (float)(tx*16)    - 20.0f;
    const float clxh = (float)(tx*16+15) + 20.0f;
    const float clyl = (float)(ty*16)    - 20.0f;
    const float clyh = (float)(ty*16+15) + 20.0f;

    for (int chunk = 0; chunk < 32; ++chunk) {
        __syncthreads();                       // prev iter done with E/P
        if (t < 16*PSTRIDE) P[t] = params[(v*NG + chunk*16)*PSTRIDE + t];
        if (chunk < 31)                        // global_prefetch_b8 next chunk
            __builtin_prefetch(&params[(v*NG + (chunk+1)*16)*PSTRIDE], 0, 0);
        __syncthreads();

        // ---- WMMA stage: E[n][halo] for 324 halo points, 16 gaussians ----
        {
            // B (6x16 coeffs, K padded to 8): lanes 0-15 rows K=0,1 | 4,5;
            // lanes 16-31 rows K=2,3 | zero pad, per 32-bit B striping.
            int pb = nl*PSTRIDE + (lo16 ? 0 : 2);
            v2f B0, B1;
            B0.x = P[pb+0];
            B0.y = P[pb+1];
            B1.x = lo16 ? P[nl*PSTRIDE+4] : 0.0f;
            B1.y = lo16 ? P[nl*PSTRIDE+5] : 0.0f;
            for (int g = wave; g < 21; g += 8) {       // 336 slots / 16 = 21 groups
                int idx = g*16 + nl;                   // 0..335 (324.. are pad)
                int hy = idx / 18, hx = idx - hy*18;
                float gy = (float)min(max(hy0 + hy, 0), 111);
                float gx = (float)min(max(hx0 + hx, 0), 111);
                // compute all features, then branchless cndmask selects
                float gxx = gx*gx, gxy = gx*gy, gyy = gy*gy;
                v2f A0, A1;                            // features of pixel M=nl
                A0.x = lo16 ? gxx : gyy;               // K0 | K2
                A0.y = lo16 ? gxy : gx;                // K1 | K3
                A1.x = lo16 ? gy : 0.0f;               // K4 | K6(pad)
                A1.y = lo16 ? 1.0f : 0.0f;             // K5 | K7(pad)
                v8f acc = {};
                acc = __builtin_amdgcn_wmma_f32_16x16x4_f32(
                          false, A0, false, B0, (short)0, acc, false, false);
                acc = __builtin_amdgcn_wmma_f32_16x16x4_f32(
                          false, A1, false, B1, (short)0, acc, false, false);
                // D: lane -> N=nl, M=r(+8 for hi half). Unconditional stores:
                // base is 16B aligned -> two ds_store_b128.
                float* dst = &E[nl*ESTR + g*16 + (lo16 ? 0 : 8)];
                v4f dlo = {acc[0], acc[1], acc[2], acc[3]};
                v4f dhi = {acc[4], acc[5], acc[6], acc[7]};
                *(v4f*)(dst)     = dlo;
                *(v4f*)(dst + 4) = dhi;
            }
        }
        __syncthreads();

        // ---- accumulate stage ----
        for (int n = 0; n < 16; ++n) {
            float xyx = P[n*PSTRIDE+6], xyy = P[n*PSTRIDE+7], wsc = P[n*PSTRIDE+8];
            // wave-uniform disc-vs-tile cull: contributes w=0 everywhere here
            if (xyx < clxl || xyx > clxh || xyy < clyl || xyy > clyh) {
#pragma unroll
                for (int i = 0; i < 4; ++i) {
                    wmn[i] = fminf(wmn[i], 0.0f); wmx[i] = fmaxf(wmx[i], 0.0f);
                }
                continue;
            }
            const float* En = &E[n*ESTR];
#pragma unroll
            for (int i = 0; i < 4; ++i) {
                float e00 = En[e00i[i]],    e01 = En[e00i[i]+1];
                float e10 = En[e00i[i]+18], e11 = En[e00i[i]+19];
                float wx = wxl_[i], wy = wyl_[i];
                float eb = wy*(wx*e00 + (1.0f-wx)*e01)
                         + (1.0f-wy)*(wx*e10 + (1.0f-wx)*e11);
                float ef = -fminf(fmaxf(eb, 0.0f), 20.0f);   // clip(-eb,-20,0)
                float dx = mkx_[i]-xyx, dy = mky_[i]-xyy;
                float w  = (dx*dx + dy*dy < 400.0f) ? wsc*__expf(ef) : 0.0f;
                S[i] += w; wmn[i] = fminf(wmn[i], w); wmx[i] = fmaxf(wmx[i], w);
            }
        }
    }
    // dm = 0.5*(S - N*min)/(max-min+eps) + S/(S+eps)
#pragma unroll
    for (int i = 0; i < 4; ++i) {
        float d = 0.5f*(S[i] - 512.0f*wmn[i]) / (wmx[i]-wmn[i]+1e-6f)
                + S[i]/(S[i]+1e-6f);
        dm[oidx[i]] = d;
    }
}

// ---------------------------------------------------------------------------
// Blur (sigma=1, r=3, zero padding) + masked merge, two passes per map.
// ---------------------------------------------------------------------------
__device__ __constant__ float KW[4] = {0.3990503f, 0.2420362f, 0.0540056f, 0.0044330f};

__global__ void blurv_k(const float* __restrict__ in, float* __restrict__ outp)
{
    int i = blockIdx.x*blockDim.x + threadIdx.x;
    if (i >= 3*FULL*FULL) return;
    int x = i % FULL, y = (i/FULL) % FULL;
    int base = i - (y*FULL + x);
    float acc = 0.0f;
#pragma unroll
    for (int d = -3; d <= 3; ++d) {
        int yy = y + d;
        if (yy >= 0 && yy < FULL) acc += KW[d < 0 ? -d : d] * in[base + yy*FULL + x];
    }
    outp[i] = acc;
}

__global__ void blurh_merge_k(const float* __restrict__ tmp, float* __restrict__ d)
{
    int i = blockIdx.x*blockDim.x + threadIdx.x;
    if (i >= 3*FULL*FULL) return;
    int x = i % FULL;
    float acc = 0.0f;
#pragma unroll
    for (int dd = -3; dd <= 3; ++dd) {
        int xx = x + dd;
        if (xx >= 0 && xx < FULL) acc += KW[dd < 0 ? -dd : dd] * tmp[i + dd];
    }
    float orig = d[i];
    d[i] = (orig > 1e-6f) ? orig : acc;
}

// ---------------------------------------------------------------------------
// Global mean / std (ddof=1) then normalize.
// ---------------------------------------------------------------------------
__global__ __launch_bounds__(1024) void stats_k(const float* __restrict__ d,
                                                float* __restrict__ stats)
{
    __shared__ float s1[1024], s2[1024];
    int t = threadIdx.x;
    float a = 0.0f, b = 0.0f;
    for (int i = t; i < 3*FULL*FULL; i += 1024) { float x = d[i]; a += x; b += x*x; }
    s1[t] = a; s2[t] = b; __syncthreads();
    for (int s = 512; s > 0; s >>= 1) {
        if (t < s) { s1[t] += s1[t+s]; s2[t] += s2[t+s]; }
        __syncthreads();
    }
    if (t == 0) {
        float n = (float)(3*FULL*FULL);
        float mean = s1[0]/n;
        float var  = (s2[0] - s1[0]*s1[0]/n) / (n - 1.0f);
        stats[0] = mean;
        stats[1] = sqrtf(fmaxf(var, 0.0f));
    }
}

__global__ void norm_k(float* __restrict__ d, const float* __restrict__ stats)
{
    int i = blockIdx.x*blockDim.x + threadIdx.x;
    if (i >= 3*FULL*FULL) return;
    d[i] = (d[i] - stats[0]) / (stats[1] + 1e-6f);
}

// ---------------------------------------------------------------------------
extern "C" void kernel_launch(void* const* d_in, const int* in_sizes, int n_in,
                              void* d_out, int out_size, void* d_ws, size_t ws_size,
                              hipStream_t stream)
{
    (void)in_sizes; (void)n_in; (void)out_size; (void)ws_size;
    const float* pos  = (const float*)d_in[0];
    const float* cov  = (const float*)d_in[1];
    const float* opa  = (const float*)d_in[2];
    const float* impo = (const float*)d_in[3];
    float* out = (float*)d_out;                 // dm maps built in place
    float* ws  = (float*)d_ws;
    float* params = ws;                         // 3*512*12 = 18432 floats
    float* tmp    = ws + 3*NG*PSTRIDE;          // 150528 floats
    float* stats  = tmp + 3*FULL*FULL;          // 2 floats

    prep_k<<<1, 512, 0, stream>>>(pos, cov, opa, impo, params);
    splat_k<<<dim3(7,7,3), 256, 0, stream>>>(params, out);

    const int NEL = 3*FULL*FULL;
    for (int step = 0; step < 2; ++step) {
        blurv_k<<<(NEL+255)/256, 256, 0, stream>>>(out, tmp);
        blurh_merge_k<<<(NEL+255)/256, 256, 0, stream>>>(tmp, out);
    }
    stats_k<<<1, 1024, 0, stream>>>(out, stats);
    norm_k<<<(NEL+255)/256, 256, 0, stream>>>(out, stats);
}